// Encoder_82042465288475
// MI455X (gfx1250) — compile-verified
//
#include <hip/hip_runtime.h>

typedef float v2f __attribute__((ext_vector_type(2)));
typedef float v8f __attribute__((ext_vector_type(8)));

#define D 128

// ---------------------------------------------------------------- zero
__global__ __launch_bounds__(256) void zero_f32(float* __restrict__ p, int n) {
  int i = blockIdx.x * blockDim.x + threadIdx.x;
  int stride = gridDim.x * blockDim.x;
  for (; i < n; i += stride) p[i] = 0.0f;
}

// ---------------------------------------------------------------- scatter
// One wave (32 lanes) per edge: lane l moves 4 floats -> 128 floats/edge.
__global__ __launch_bounds__(256) void scatter_accum(
    const float* __restrict__ x, const int* __restrict__ src,
    const int* __restrict__ dst, float* __restrict__ agg,
    float* __restrict__ cnt, int E) {
  int wave = threadIdx.x >> 5;
  int lane = threadIdx.x & 31;
  int e = blockIdx.x * 8 + wave;
  if (e >= E) return;
  int s = src[e];
  int d = dst[e];
  const float4* xs = (const float4*)(x + (long long)s * D);
  float4 v = xs[lane];
  float* ag = agg + (long long)d * D + lane * 4;
  atomicAdd(ag + 0, v.x);
  atomicAdd(ag + 1, v.y);
  atomicAdd(ag + 2, v.z);
  atomicAdd(ag + 3, v.w);
  if (lane == 0) atomicAdd(cnt + d, 1.0f);
}

// ---------------------------------------------------------------- mean
__global__ __launch_bounds__(256) void finalize_mean(
    float* __restrict__ agg, const float* __restrict__ cnt, int total) {
  int i = blockIdx.x * blockDim.x + threadIdx.x;
  int stride = gridDim.x * blockDim.x;
  for (; i < total; i += stride) {
    float c = cnt[i >> 7];              // row = i / 128
    agg[i] = agg[i] / fmaxf(c, 1.0f);
  }
}

// ---------------------------------------------------------------- fused GEMM
// out[M,128] = A1 @ Wl + A2 @ Wr + bias   (optional ReLU)
// One wave computes one 16x16 output tile via V_WMMA_F32_16X16X4_F32.
// A (16x4 f32):  lane&15 = row M, lanes>=16 hold K+2/K+3 halves.
// B (4x16 f32):  lane&15 = col N, lanes>=16 hold K+2/K+3 halves.
// C/D (16x16):   VGPR v, lane<16 -> row m0+v, lane>=16 -> row m0+v+8.
__global__ __launch_bounds__(256) void sage_gemm(
    const float* __restrict__ A1, const float* __restrict__ A2,
    const float* __restrict__ Wl, const float* __restrict__ Wr,
    const float* __restrict__ bias, float* __restrict__ out,
    int M, int doRelu) {
  int wave = threadIdx.x >> 5;
  int lane = threadIdx.x & 31;
  int t = blockIdx.x * 8 + wave;          // tile id
  int m0 = (t >> 3) << 4;                 // 8 col-strips of 16 over HID=128
  int n0 = (t & 7) << 4;
  if (m0 >= M) return;                    // wave-uniform exit

  int r = lane & 15;                      // A row within tile / B col within tile
  int koff = (lane >> 4) << 1;            // 0 for lanes 0-15, 2 for lanes 16-31

  v8f acc = {};
  const float* a1row = A1 + (long long)(m0 + r) * D;
  const float* a2row = A2 + (long long)(m0 + r) * D;

#pragma unroll
  for (int kb = 0; kb < D; kb += 4) {
    v2f a, b;
    a.x = a1row[kb + koff];
    a.y = a1row[kb + koff + 1];
    b.x = Wl[(kb + koff) * D + n0 + r];
    b.y = Wl[(kb + koff + 1) * D + n0 + r];
    acc = __builtin_amdgcn_wmma_f32_16x16x4_f32(false, a, false, b,
                                                (short)0, acc, false, false);
  }
#pragma unroll
  for (int kb = 0; kb < D; kb += 4) {
    v2f a, b;
    a.x = a2row[kb + koff];
    a.y = a2row[kb + koff + 1];
    b.x = Wr[(kb + koff) * D + n0 + r];
    b.y = Wr[(kb + koff + 1) * D + n0 + r];
    acc = __builtin_amdgcn_wmma_f32_16x16x4_f32(false, a, false, b,
                                                (short)0, acc, false, false);
  }

  float bv = bias[n0 + r];
  int rowAdd = (lane >> 4) << 3;          // +8 rows for upper half-wave
#pragma unroll
  for (int v = 0; v < 8; ++v) {
    float val = acc[v] + bv;
    if (doRelu) val = fmaxf(val, 0.0f);
    out[(long long)(m0 + v + rowAdd) * D + n0 + r] = val;
  }
}

// ---------------------------------------------------------------- launch
extern "C" void kernel_launch(void* const* d_in, const int* in_sizes, int n_in,
                              void* d_out, int out_size, void* d_ws, size_t ws_size,
                              hipStream_t stream) {
  const float* x   = (const float*)d_in[0];
  const int*   ei  = (const int*)d_in[1];
  const float* W1l = (const float*)d_in[2];
  const float* W1r = (const float*)d_in[3];
  const float* b1  = (const float*)d_in[4];
  const float* W2l = (const float*)d_in[5];
  const float* W2r = (const float*)d_in[6];
  const float* b2  = (const float*)d_in[7];

  const int N = in_sizes[0] / D;   // 50000
  const int E = in_sizes[1] / 2;   // 625000
  const int* src = ei;
  const int* dst = ei + E;

  float* agg = (float*)d_ws;                  // N*128 floats
  float* cnt = agg + (size_t)N * D;           // N floats
  float* h   = cnt + N;                       // N*128 floats
  float* out = (float*)d_out;

  const int zn = N * (D + 1);                 // agg + cnt are contiguous
  int zblk = (zn + 255) / 256; if (zblk > 4096) zblk = 4096;
  int fblk = (N * D + 255) / 256; if (fblk > 4096) fblk = 4096;
  const int eblk = (E + 7) / 8;               // 8 edges (waves) per block
  const int mt = (N + 15) / 16;
  const int gblk = (mt * 8 + 7) / 8;          // 8 tiles (waves) per block

  // ---- layer 1 ----
  zero_f32<<<zblk, 256, 0, stream>>>(agg, zn);
  scatter_accum<<<eblk, 256, 0, stream>>>(x, src, dst, agg, cnt, E);
  finalize_mean<<<fblk, 256, 0, stream>>>(agg, cnt, N * D);
  sage_gemm<<<gblk, 256, 0, stream>>>(agg, x, W1l, W1r, b1, h, N, /*relu=*/1);

  // ---- layer 2 ----
  zero_f32<<<zblk, 256, 0, stream>>>(agg, zn);
  scatter_accum<<<eblk, 256, 0, stream>>>(h, src, dst, agg, cnt, E);
  finalize_mean<<<fblk, 256, 0, stream>>>(agg, cnt, N * D);
  sage_gemm<<<gblk, 256, 0, stream>>>(agg, h, W2l, W2r, b2, out, N, /*relu=*/0);
}